// RelativePositionalEmbedding2D_57887569215531
// MI455X (gfx1250) — compile-verified
//
#include <hip/hip_runtime.h>

typedef __attribute__((ext_vector_type(2))) float v2f;
typedef __attribute__((ext_vector_type(4))) float v4f;
typedef __attribute__((ext_vector_type(8))) float v8f;

// Problem constants (reference: B=2, HEADS=8, T=8, QH=QW=16, D=64; K* == Q*)
#define NB 2
#define NH 8
#define NT 8
#define QH 16
#define QW 16
#define DD 64
#define NROWS (NB * NH * NT * QH * QW)   // 32768 query rows
#define KROW  (NT * QH * QW)             // 2048 key positions per row

// ---------------------------------------------------------------------------
// Phase 1a: rel_h and rel_t via V_WMMA_F32_16X16X4_F32.
// One wave per (b,N,t,h) tile. A = q[b,N,t,h, w=0..15, :] (16x64 contiguous).
//   rel_h[...,w,kh] = sum_c A[w,c] * Hemb[(h-kh+15), c]   -> B[c][kh]
//   rel_t[...,w,kt] = sum_c A[w,c] * Temb[(t-kt+7),  c]   -> B[c][kt], kt<8
// ---------------------------------------------------------------------------
__global__ __launch_bounds__(32) void bias_ht_kernel(
    const float* __restrict__ q, const float* __restrict__ hemb,
    const float* __restrict__ temb, float* __restrict__ relh,
    float* __restrict__ relt) {
  const int tile = blockIdx.x;        // 0 .. 2047
  const int h    = tile & 15;
  const int bNt  = tile >> 4;         // flattened (b, N, t)
  const int t    = bNt & 7;

  const int lane  = threadIdx.x;      // 0..31 (wave32)
  const int m     = lane & 15;        // A row (w) / B column (kh or kt)
  const int khalf = (lane >> 4) * 2;  // K sub-offset within a 4-step

  const float* arow = q + (((size_t)bNt * QH + h) * QW + m) * DD + khalf;
  const float* brh  = hemb + (size_t)(h - m + 15) * DD + khalf;
  const float* brt  = temb + (size_t)(t - (m & 7) + 7) * DD + khalf; // safe addr

  v8f acc_h = {};
  v8f acc_t = {};
#pragma unroll
  for (int s = 0; s < 16; ++s) {
    v2f a  = *(const v2f*)(arow + s * 4);
    v2f bh = *(const v2f*)(brh + s * 4);
    v2f bt = {0.f, 0.f};
    if (m < 8) bt = *(const v2f*)(brt + s * 4);  // zero-pad kt columns 8..15
    acc_h = __builtin_amdgcn_wmma_f32_16x16x4_f32(false, a, false, bh,
                                                  (short)0, acc_h, false, false);
    acc_t = __builtin_amdgcn_wmma_f32_16x16x4_f32(false, a, false, bt,
                                                  (short)0, acc_t, false, false);
  }

  // C/D layout: vgpr i -> M = i + 8*(lane/16); N = lane%16
  const int mbase = (lane >> 4) * 8;
  const int n     = m;
  float* ph = relh + ((size_t)bNt * QH + h) * QW * 16;  // [w, kh]
  float* pt = relt + ((size_t)bNt * QH + h) * QW * 8;   // [w, kt]
#pragma unroll
  for (int i = 0; i < 8; ++i) {
    const int M = mbase + i;
    ph[M * 16 + n] = acc_h[i];
    if (n < 8) pt[M * 8 + n] = acc_t[i];
  }
}

// ---------------------------------------------------------------------------
// Phase 1b: rel_w. One wave per (b,N,t,w); A rows run over h (stride QW*DD).
//   rel_w[...,h,w,kw] = sum_c q[b,N,t,h,w,c] * Wemb[(w-kw+15), c]
// ---------------------------------------------------------------------------
__global__ __launch_bounds__(32) void bias_w_kernel(
    const float* __restrict__ q, const float* __restrict__ wemb,
    float* __restrict__ relw) {
  const int tile = blockIdx.x;        // 0 .. 2047
  const int w    = tile & 15;
  const int bNt  = tile >> 4;

  const int lane  = threadIdx.x;
  const int m     = lane & 15;        // A row (h) / B column (kw)
  const int khalf = (lane >> 4) * 2;

  const float* arow = q + (((size_t)bNt * QH + m) * QW + w) * DD + khalf;
  const float* brw  = wemb + (size_t)(w - m + 15) * DD + khalf;

  v8f acc = {};
#pragma unroll
  for (int s = 0; s < 16; ++s) {
    v2f a = *(const v2f*)(arow + s * 4);
    v2f b = *(const v2f*)(brw + s * 4);
    acc = __builtin_amdgcn_wmma_f32_16x16x4_f32(false, a, false, b,
                                                (short)0, acc, false, false);
  }

  const int mbase = (lane >> 4) * 8;
  const int n     = m;
#pragma unroll
  for (int i = 0; i < 8; ++i) {
    const int M = mbase + i;  // h
    relw[(((size_t)bNt * QH + M) * QW + w) * 16 + n] = acc[i];
  }
}

// ---------------------------------------------------------------------------
// Phase 2: the bandwidth phase. One block per query row: stream 2048 floats
// of scores -> out with float4 NT loads/stores, adding the 3 broadcast biases.
// ---------------------------------------------------------------------------
__global__ __launch_bounds__(256) void add_bias_kernel(
    const float* __restrict__ scores, const float* __restrict__ relh,
    const float* __restrict__ relw, const float* __restrict__ relt,
    float* __restrict__ out) {
  const size_t r = blockIdx.x;  // 0 .. NROWS-1, matches [B,H,T,h,w] flattening
  __shared__ float sh[16];
  __shared__ float sw[16];
  __shared__ float st[8];

  const int tid = threadIdx.x;
  if (tid < 16)       sh[tid]      = relh[r * 16 + tid];
  else if (tid < 32)  sw[tid - 16] = relw[r * 16 + (tid - 16)];
  else if (tid < 40)  st[tid - 32] = relt[r * 8 + (tid - 32)];
  __syncthreads();

  const float* sp = scores + r * KROW;
  float* op       = out + r * KROW;

  const int e0 = tid * 8;  // 256 threads * 8 floats = 2048
#pragma unroll
  for (int j = 0; j < 2; ++j) {
    const int e  = e0 + j * 4;
    const int kw = e & 15;
    const int kh = (e >> 4) & 15;
    const int kt = e >> 8;
    v4f v = __builtin_nontemporal_load((const v4f*)(sp + e));
    const float c = st[kt] + sh[kh];
    v.x += c + sw[kw + 0];
    v.y += c + sw[kw + 1];
    v.z += c + sw[kw + 2];
    v.w += c + sw[kw + 3];
    __builtin_nontemporal_store(v, (v4f*)(op + e));
  }
}

extern "C" void kernel_launch(void* const* d_in, const int* in_sizes, int n_in,
                              void* d_out, int out_size, void* d_ws,
                              size_t ws_size, hipStream_t stream) {
  const float* query  = (const float*)d_in[0];
  const float* scores = (const float*)d_in[1];
  const float* hemb   = (const float*)d_in[2];
  const float* wemb   = (const float*)d_in[3];
  const float* temb   = (const float*)d_in[4];
  float* out = (float*)d_out;

  float* relh = (float*)d_ws;              // NROWS*16 floats (2 MB)
  float* relw = relh + (size_t)NROWS * 16; // NROWS*16 floats (2 MB)
  float* relt = relw + (size_t)NROWS * 16; // NROWS*8  floats (1 MB)

  bias_ht_kernel<<<NB * NH * NT * QH, 32, 0, stream>>>(query, hemb, temb,
                                                       relh, relt);
  bias_w_kernel<<<NB * NH * NT * QW, 32, 0, stream>>>(query, wemb, relw);
  add_bias_kernel<<<NROWS, 256, 0, stream>>>(scores, relh, relw, relt, out);
}